// GAT_33973191311454
// MI455X (gfx1250) — compile-verified
//
#include <hip/hip_runtime.h>
#include <hip/hip_bf16.h>

#define NN 50000
#define EE 800000
#define KDIM 256
#define OUTD 64
#define NEG_SLOPE 0.2f

typedef __attribute__((ext_vector_type(2))) float v2f;
typedef __attribute__((ext_vector_type(4))) float v4f;
typedef __attribute__((ext_vector_type(8))) float v8f;

__device__ __forceinline__ unsigned orderFloat(float f) {
  unsigned u = __float_as_uint(f);
  return (u & 0x80000000u) ? ~u : (u | 0x80000000u);
}
__device__ __forceinline__ float unorderFloat(unsigned u) {
  u = (u & 0x80000000u) ? (u & 0x7FFFFFFFu) : ~u;
  return __uint_as_float(u);
}

// ---------------------------------------------------------------------------
// init: rst <- gat_bias (folds bias into the scatter accumulator);
//       m (ordered-uint) <- 0 (== "less than any real score"); denom <- 0
// ---------------------------------------------------------------------------
__global__ __launch_bounds__(256) void init_kernel(
    float* __restrict__ rst, const float* __restrict__ gat_bias,
    unsigned* __restrict__ m_u32, float* __restrict__ denom) {
  int i = blockIdx.x * blockDim.x + threadIdx.x;
  if (i < NN * KDIM) rst[i] = gat_bias[i & (KDIM - 1)];
  if (i < NN * 4) { m_u32[i] = 0u; denom[i] = 0.0f; }
}

// ---------------------------------------------------------------------------
// Projection GEMM: H = feat @ W + b, for W in {w_src, w_dst}.
// One wave computes a 64-row x 16-col tile with v_wmma_f32_16x16x4_f32.
// A frag (16x4 f32): lane = m + 16*(k>>1), vgpr = k&1
// B frag (4x16 f32): lane = n + 16*(k>>1), vgpr = k&1
// D (16x16 f32):     lane = n + 16*(m>=8), vgpr = m&7
// ---------------------------------------------------------------------------
__global__ __launch_bounds__(32) void wmma_proj_kernel(
    const float* __restrict__ feat,
    const float* __restrict__ Wsrc, const float* __restrict__ Bsrc,
    const float* __restrict__ Wdst, const float* __restrict__ Bdst,
    float* __restrict__ Hsrc, float* __restrict__ Hdst) {
  const int lane = threadIdx.x;
  const int rowTile = blockIdx.x;          // 64-row group
  const int yt = blockIdx.y;               // 0..31
  const int mat = yt >> 4;                 // 0 = src, 1 = dst
  const int colTile = yt & 15;             // 16-col group

  const float* __restrict__ W  = mat ? Wdst : Wsrc;
  const float* __restrict__ Bv = mat ? Bdst : Bsrc;
  float* __restrict__ Hout     = mat ? Hdst : Hsrc;

  const int row0 = rowTile * 64;
  const int col0 = colTile * 16;
  const int khi  = lane >> 4;              // 0/1 -> K pair selector
  const int mlo  = lane & 15;              // m (A) / n (B,D)

  int arow[4];
#pragma unroll
  for (int s = 0; s < 4; ++s) {
    int r = row0 + s * 16 + mlo;
    arow[s] = (r < NN) ? r : 0;            // clamp; masked at store
  }

  v8f acc[4] = {};
  for (int kb = 0; kb < KDIM; kb += 4) {
    v2f bfrag;
    bfrag.x = W[(size_t)(kb + 2 * khi)     * KDIM + col0 + mlo];
    bfrag.y = W[(size_t)(kb + 2 * khi + 1) * KDIM + col0 + mlo];
#pragma unroll
    for (int s = 0; s < 4; ++s) {
      v2f afrag = *(const v2f*)(feat + (size_t)arow[s] * KDIM + kb + 2 * khi);
      acc[s] = __builtin_amdgcn_wmma_f32_16x16x4_f32(
          false, afrag, false, bfrag, (short)0, acc[s], false, false);
    }
  }

  const int mbase = khi * 8;
  const int ncol  = col0 + mlo;
  const float bias = Bv[ncol];
  if (row0 + 64 <= NN) {
    // whole tile in range: unpredicated stores (781 of 782 row tiles)
#pragma unroll
    for (int s = 0; s < 4; ++s) {
#pragma unroll
      for (int i = 0; i < 8; ++i) {
        int r = row0 + s * 16 + mbase + i;
        Hout[(size_t)r * KDIM + ncol] = acc[s][i] + bias;
      }
    }
  } else {
#pragma unroll
    for (int s = 0; s < 4; ++s) {
#pragma unroll
      for (int i = 0; i < 8; ++i) {
        int r = row0 + s * 16 + mbase + i;
        if (r < NN) Hout[(size_t)r * KDIM + ncol] = acc[s][i] + bias;
      }
    }
  }
}

// ---------------------------------------------------------------------------
// Edge score: wave per edge. Lane l handles 8 contiguous (h,d) elements,
// so lanes [8h, 8h+7] cover head h. score = a . leaky_relu(hs + hd).
// Segment-max via atomicMax on monotonic uint encoding of float.
// ---------------------------------------------------------------------------
__global__ __launch_bounds__(256) void edge_score_kernel(
    const int* __restrict__ ei, const float* __restrict__ Hsrc,
    const float* __restrict__ Hdst, const float* __restrict__ attn,
    float* __restrict__ score, unsigned* __restrict__ m_u32) {
  const int e = (blockIdx.x * blockDim.x + threadIdx.x) >> 5;
  const int lane = threadIdx.x & 31;
  if (e >= EE) return;
  const int src = ei[e];
  const int dst = ei[EE + e];

  const v4f* ps = (const v4f*)(Hsrc + (size_t)src * KDIM + lane * 8);
  const v4f* pd = (const v4f*)(Hdst + (size_t)dst * KDIM + lane * 8);
  const v4f* pa = (const v4f*)(attn + lane * 8);
  v4f s0 = ps[0], s1 = ps[1];
  v4f d0 = pd[0], d1 = pd[1];
  v4f a0 = pa[0], a1 = pa[1];

  float part = 0.0f;
#pragma unroll
  for (int i = 0; i < 4; ++i) {
    float v = s0[i] + d0[i];
    v = (v >= 0.0f) ? v : v * NEG_SLOPE;
    part += v * a0[i];
    float w = s1[i] + d1[i];
    w = (w >= 0.0f) ? w : w * NEG_SLOPE;
    part += w * a1[i];
  }
  part += __shfl_xor(part, 1, 32);
  part += __shfl_xor(part, 2, 32);
  part += __shfl_xor(part, 4, 32);
  if ((lane & 7) == 0) {
    int h = lane >> 3;
    score[(size_t)e * 4 + h] = part;
    atomicMax(&m_u32[(size_t)dst * 4 + h], orderFloat(part));
  }
}

// ---------------------------------------------------------------------------
// m <- isfinite(decode(m)) ? decode(m) : 0   (in place: uint -> float)
// ---------------------------------------------------------------------------
__global__ __launch_bounds__(256) void sanitize_m_kernel(unsigned* __restrict__ m) {
  int i = blockIdx.x * blockDim.x + threadIdx.x;
  if (i >= NN * 4) return;
  float f = unorderFloat(m[i]);
  if (!isfinite(f)) f = 0.0f;
  ((float*)m)[i] = f;
}

// ---------------------------------------------------------------------------
// ex = exp(score - m[dst]); denom[dst] += ex   (thread per edge*head)
// ---------------------------------------------------------------------------
__global__ __launch_bounds__(256) void edge_exp_kernel(
    const int* __restrict__ ei, const float* __restrict__ m_f,
    float* __restrict__ score, float* __restrict__ denom) {
  int i = blockIdx.x * blockDim.x + threadIdx.x;
  if (i >= EE * 4) return;
  int e = i >> 2, h = i & 3;
  int dst = ei[EE + e];
  float ex = expf(score[i] - m_f[(size_t)dst * 4 + h]);
  score[i] = ex;
  unsafeAtomicAdd(&denom[(size_t)dst * 4 + h], ex);
}

// ---------------------------------------------------------------------------
// Scatter: rst[dst] += alpha * Hsrc[src]. Wave per edge, 8 f32 atomics/lane.
// ---------------------------------------------------------------------------
__global__ __launch_bounds__(256) void edge_scatter_kernel(
    const int* __restrict__ ei, const float* __restrict__ Hsrc,
    const float* __restrict__ ex, const float* __restrict__ denom,
    float* __restrict__ rst) {
  const int e = (blockIdx.x * blockDim.x + threadIdx.x) >> 5;
  const int lane = threadIdx.x & 31;
  if (e >= EE) return;
  const int src = ei[e];
  const int dst = ei[EE + e];
  const int h = lane >> 3;
  float den = denom[(size_t)dst * 4 + h];
  float alpha = ex[(size_t)e * 4 + h] / ((den == 0.0f) ? 1.0f : den);
  const v4f* ps = (const v4f*)(Hsrc + (size_t)src * KDIM + lane * 8);
  v4f s0 = ps[0], s1 = ps[1];
  float* pr = rst + (size_t)dst * KDIM + lane * 8;
#pragma unroll
  for (int i = 0; i < 4; ++i) {
    unsafeAtomicAdd(&pr[i],     alpha * s0[i]);
    unsafeAtomicAdd(&pr[i + 4], alpha * s1[i]);
  }
}

// ---------------------------------------------------------------------------
// Output GEMM: z = rst @ w_fc + b_fc  (rst already includes gat_bias).
// Same 64x16-per-wave fp32 WMMA tiling; 4 column tiles (OUTD=64).
// ---------------------------------------------------------------------------
__global__ __launch_bounds__(32) void wmma_out_kernel(
    const float* __restrict__ rst, const float* __restrict__ Wfc,
    const float* __restrict__ bfc, float* __restrict__ out) {
  const int lane = threadIdx.x;
  const int row0 = blockIdx.x * 64;
  const int col0 = blockIdx.y * 16;
  const int khi = lane >> 4;
  const int mlo = lane & 15;

  int arow[4];
#pragma unroll
  for (int s = 0; s < 4; ++s) {
    int r = row0 + s * 16 + mlo;
    arow[s] = (r < NN) ? r : 0;
  }

  v8f acc[4] = {};
  for (int kb = 0; kb < KDIM; kb += 4) {
    v2f bfrag;
    bfrag.x = Wfc[(size_t)(kb + 2 * khi)     * OUTD + col0 + mlo];
    bfrag.y = Wfc[(size_t)(kb + 2 * khi + 1) * OUTD + col0 + mlo];
#pragma unroll
    for (int s = 0; s < 4; ++s) {
      v2f afrag = *(const v2f*)(rst + (size_t)arow[s] * KDIM + kb + 2 * khi);
      acc[s] = __builtin_amdgcn_wmma_f32_16x16x4_f32(
          false, afrag, false, bfrag, (short)0, acc[s], false, false);
    }
  }

  const int mbase = khi * 8;
  const int ncol  = col0 + mlo;
  const float bias = bfc[ncol];
  if (row0 + 64 <= NN) {
#pragma unroll
    for (int s = 0; s < 4; ++s) {
#pragma unroll
      for (int i = 0; i < 8; ++i) {
        int r = row0 + s * 16 + mbase + i;
        out[(size_t)r * OUTD + ncol] = acc[s][i] + bias;
      }
    }
  } else {
#pragma unroll
    for (int s = 0; s < 4; ++s) {
#pragma unroll
      for (int i = 0; i < 8; ++i) {
        int r = row0 + s * 16 + mbase + i;
        if (r < NN) out[(size_t)r * OUTD + ncol] = acc[s][i] + bias;
      }
    }
  }
}

extern "C" void kernel_launch(void* const* d_in, const int* in_sizes, int n_in,
                              void* d_out, int out_size, void* d_ws, size_t ws_size,
                              hipStream_t stream) {
  const float* feat     = (const float*)d_in[0];
  const int*   ei       = (const int*)d_in[1];
  const float* w_src    = (const float*)d_in[2];
  const float* b_src    = (const float*)d_in[3];
  const float* w_dst    = (const float*)d_in[4];
  const float* b_dst    = (const float*)d_in[5];
  const float* attn     = (const float*)d_in[6];
  const float* gat_bias = (const float*)d_in[7];
  const float* w_fc     = (const float*)d_in[8];
  const float* b_fc     = (const float*)d_in[9];
  float* out = (float*)d_out;

  // Workspace layout (floats): Hsrc | Hdst | score/ex | m | denom | rst
  float* ws    = (float*)d_ws;
  float* Hsrc  = ws;                        // N*256
  float* Hdst  = Hsrc + (size_t)NN * KDIM;  // N*256
  float* score = Hdst + (size_t)NN * KDIM;  // E*4
  float* mbuf  = score + (size_t)EE * 4;    // N*4 (uint then float)
  float* denom = mbuf + (size_t)NN * 4;     // N*4
  float* rst   = denom + (size_t)NN * 4;    // N*256

  const int rowTiles = (NN + 63) / 64;      // 782

  init_kernel<<<(NN * KDIM + 255) / 256, 256, 0, stream>>>(
      rst, gat_bias, (unsigned*)mbuf, denom);

  wmma_proj_kernel<<<dim3(rowTiles, 32), 32, 0, stream>>>(
      feat, w_src, b_src, w_dst, b_dst, Hsrc, Hdst);

  edge_score_kernel<<<(EE * 32) / 256, 256, 0, stream>>>(
      ei, Hsrc, Hdst, attn, score, (unsigned*)mbuf);

  sanitize_m_kernel<<<(NN * 4 + 255) / 256, 256, 0, stream>>>((unsigned*)mbuf);

  edge_exp_kernel<<<(EE * 4 + 255) / 256, 256, 0, stream>>>(
      ei, mbuf, score, denom);

  edge_scatter_kernel<<<(EE * 32) / 256, 256, 0, stream>>>(
      ei, Hsrc, score, denom, rst);

  wmma_out_kernel<<<dim3(rowTiles, 4), 32, 0, stream>>>(rst, w_fc, b_fc, out);
}